// ARM_71665824301873
// MI455X (gfx1250) — compile-verified
//
#include <hip/hip_runtime.h>

typedef __attribute__((ext_vector_type(2))) float v2f;
typedef __attribute__((ext_vector_type(8))) float v8f;

#define HW    64
#define DPIX  4096      // 64*64 pixels per image
#define KCLS  256

// One wave computes 16 pixels x 256 classes via 16 tiles of V_WMMA_F32_16X16X4_F32
// (GEMM-K padded 9 -> 16, i.e. 4 chained K=4 WMMAs per tile), then does the
// softmax + gather + NLL entirely in registers with wave32 shuffles.
__global__ __launch_bounds__(256) void ARM_conv_nll_wmma(
    const float* __restrict__ x,      // (64,1,64,64)
    const float* __restrict__ w,      // (256,1,3,3)
    const float* __restrict__ bias,   // (256,)
    float* __restrict__ partial) {    // per-block partial sums of -log p
  __shared__ float Wlds[KCLS * 16];   // B panel: [class][kk], kk 9..15 zero
  __shared__ float wsum[8];

  const int tid = threadIdx.x;

  // Stage weights into LDS, zero-padded along GEMM-K (one class per thread).
  {
    const int c = tid;
#pragma unroll
    for (int kk = 0; kk < 16; ++kk)
      Wlds[c * 16 + kk] = (kk < 9) ? w[c * 9 + kk] : 0.0f;
  }
  __syncthreads();

  const int lane = tid & 31;
  const int wave = tid >> 5;
  const int half = lane >> 4;   // which 16-lane half
  const int mcol = lane & 15;   // A: M row / B,C: N column
  const int base = (blockIdx.x * 8 + wave) * 16;   // first pixel of M-tile

  // ---- A fragments (im2col rows, K = 3x3 window zero-padded to 16) ----
  const int p   = base + mcol;       // this lane's A-matrix pixel
  const int n   = p >> 12;
  const int rem = p & 4095;
  const int y   = rem >> 6;
  const int xc  = rem & 63;
  const float* xn = x + (n << 12);

  v2f afrag[4];
#pragma unroll
  for (int s = 0; s < 4; ++s) {
#pragma unroll
    for (int j = 0; j < 2; ++j) {
      const int kk = 4 * s + 2 * half + j;   // GEMM-K index held by this lane
      float val = 0.0f;
      if (kk < 9) {
        const int yy = y + kk / 3 - 1;
        const int xx = xc + kk % 3 - 1;
        if (yy >= 0 && yy < HW && xx >= 0 && xx < HW)
          val = xn[(yy << 6) + xx];
      }
      afrag[s][j] = val;
    }
  }

  // ---- GEMM: 16 class-tiles, 4 chained f32 WMMAs each (bias pre-loaded) ----
  v8f acc[16];
#pragma unroll
  for (int t = 0; t < 16; ++t) {
    const float bv = bias[16 * t + mcol];
    v8f c = {bv, bv, bv, bv, bv, bv, bv, bv};
#pragma unroll
    for (int s = 0; s < 4; ++s) {
      const v2f bfrag =
          *(const v2f*)&Wlds[(16 * t + mcol) * 16 + 4 * s + 2 * half];
      c = __builtin_amdgcn_wmma_f32_16x16x4_f32(
          false, afrag[s], false, bfrag, (short)0, c, false, false);
    }
    acc[t] = c;
  }

  // ---- Softmax over 256 classes + gather + clamped log-prob, per pixel ----
  const float LOGEPS   = -11.512925465f;        // log(1e-5)
  const float LOG1MEPS = -1.0000050000e-05f;    // log(1 - 1e-5)
  float nll = 0.0f;

#pragma unroll
  for (int v = 0; v < 8; ++v) {
    // pixel owned by this half at C-row v:  M = v + 8*half
    const int pm = base + v + 8 * half;

    // row max over 16 tiles then across the 16-lane half
    float m = acc[0][v];
#pragma unroll
    for (int t = 1; t < 16; ++t) m = fmaxf(m, acc[t][v]);
#pragma unroll
    for (int msk = 1; msk <= 8; msk <<= 1)
      m = fmaxf(m, __shfl_xor(m, msk, 32));

    // sum of exp
    float s = 0.0f;
#pragma unroll
    for (int t = 0; t < 16; ++t) s += expf(acc[t][v] - m);
#pragma unroll
    for (int msk = 1; msk <= 8; msk <<= 1)
      s += __shfl_xor(s, msk, 32);
    const float lse = m + logf(s);

    // gather logit at class idx = int(x * 255)
    const float xval = x[pm];                 // same for all lanes in the half
    const int idx = (int)(xval * 255.0f);
    const int tt = idx >> 4, nn = idx & 15;
    float g = 0.0f;
#pragma unroll
    for (int t = 0; t < 16; ++t)
      g += (t == tt && mcol == nn) ? acc[t][v] : 0.0f;
#pragma unroll
    for (int msk = 1; msk <= 8; msk <<= 1)
      g += __shfl_xor(g, msk, 32);

    float lp = g - lse;                       // log softmax at idx
    lp = fminf(fmaxf(lp, LOGEPS), LOG1MEPS);  // == log(clip(prob, eps, 1-eps))
    if (mcol == 0) nll += -lp;                // count each pixel once
  }

  // wave reduction -> block reduction -> per-block partial
#pragma unroll
  for (int msk = 1; msk <= 16; msk <<= 1)
    nll += __shfl_xor(nll, msk, 32);
  if (lane == 0) wsum[wave] = nll;
  __syncthreads();
  if (tid == 0) {
    float t = 0.0f;
#pragma unroll
    for (int i = 0; i < 8; ++i) t += wsum[i];
    partial[blockIdx.x] = t;
  }
}

__global__ __launch_bounds__(256) void ARM_reduce_mean(
    const float* __restrict__ partial, float* __restrict__ out, int nblocks) {
  __shared__ float smem[256];
  float a = 0.0f;
  for (int i = threadIdx.x; i < nblocks; i += 256) a += partial[i];
  smem[threadIdx.x] = a;
  __syncthreads();
  for (int st = 128; st > 0; st >>= 1) {
    if (threadIdx.x < st) smem[threadIdx.x] += smem[threadIdx.x + st];
    __syncthreads();
  }
  if (threadIdx.x == 0) out[0] = smem[0] * (1.0f / 64.0f);  // mean over N=64
}

extern "C" void kernel_launch(void* const* d_in, const int* in_sizes, int n_in,
                              void* d_out, int out_size, void* d_ws, size_t ws_size,
                              hipStream_t stream) {
  const float* x = (const float*)d_in[0];   // 64*1*64*64
  const float* w = (const float*)d_in[1];   // 256*1*3*3
  const float* b = (const float*)d_in[2];   // 256
  float* out = (float*)d_out;
  float* partial = (float*)d_ws;            // 2048 floats of scratch

  const int total_pixels = 64 * DPIX;       // 262144
  const int nblocks = total_pixels / 128;   // 128 pixels / block (8 waves x 16)

  ARM_conv_nll_wmma<<<nblocks, 256, 0, stream>>>(x, w, b, partial);
  ARM_reduce_mean<<<1, 256, 0, stream>>>(partial, out, nblocks);
}